// PulseRetrievalLossFunction_9019431322312
// MI455X (gfx1250) — compile-verified
//
#include <hip/hip_runtime.h>

// ---------------------------------------------------------------------------
// FROG pulse-retrieval loss on MI455X (gfx1250, wave32).
// DFT-as-GEMM with V_WMMA_F32_16X16X4_F32; each wave computes a 16x32 output
// strip (two 16x16 WMMA tiles sharing one A fragment) with fused magnitude
// and row/col/dot/sum-sq reductions. Tref is never materialized.
// ---------------------------------------------------------------------------

#define NN 1024
#define TS_F 1.5f
#define W_CENTER_F 337.927f

typedef __attribute__((ext_vector_type(2))) float v2f;
typedef __attribute__((ext_vector_type(8))) float v8f;

// --------------------------- zero accumulators -----------------------------
__global__ void zero_kernel(float* __restrict__ p, int n) {
    int i = blockIdx.x * blockDim.x + threadIdx.x;
    if (i < n) p[i] = 0.0f;
}

// --------------------------- fill DFT matrix G -----------------------------
// G[j,k] = TS * exp(-2*pi*i*j*k/N) * (-1)^(j+k)   (fftshifts folded in)
__global__ __launch_bounds__(256) void fill_G(float* __restrict__ Gr,
                                              float* __restrict__ Gi) {
    int idx = blockIdx.x * 256 + threadIdx.x;       // 0 .. N*N-1
    int j = idx >> 10;
    int k = idx & (NN - 1);
    int r = (j * k) & (NN - 1);                     // jk mod N (exact, small arg)
    float th = (float)r * (6.283185307179586f / (float)NN);
    float s, c;
    sincosf(th, &s, &c);
    float sign = ((j + k) & 1) ? -TS_F : TS_F;
    Gr[idx] = sign * c;    // Re(W^{jk}) = cos
    Gi[idx] = -sign * s;   // Im(W^{jk}) = -sin
}

// ------------------- spectrogram statistics (Tmeas) ------------------------
// colsumTm[k] = sum_m sp[m,k];  sumSq += sum sp^2;  tmax = max sp (>=0)
__global__ __launch_bounds__(256) void spectro_stats(const float* __restrict__ sp,
                                                     float* __restrict__ colsumTm,
                                                     float* __restrict__ sumSq,
                                                     unsigned int* __restrict__ tmaxBits) {
    int k = blockIdx.x * 256 + threadIdx.x;         // grid = N/256 blocks
    float cs = 0.f, sq = 0.f, mx = 0.f;
    for (int m = 0; m < NN; ++m) {
        float v = sp[m * NN + k];                   // consecutive lanes -> coalesced
        cs += v; sq += v * v; mx = fmaxf(mx, v);
    }
    colsumTm[k] = cs;
    for (int msk = 1; msk < 32; msk <<= 1) {
        sq += __shfl_xor(sq, msk, 32);
        mx = fmaxf(mx, __shfl_xor(mx, msk, 32));
    }
    __shared__ float ssq[8];
    __shared__ float smx[8];
    int lane = threadIdx.x & 31, wave = threadIdx.x >> 5;
    if (lane == 0) { ssq[wave] = sq; smx[wave] = mx; }
    __syncthreads();
    if (threadIdx.x == 0) {
        float a = 0.f, b = 0.f;
        for (int w = 0; w < 8; ++w) { a += ssq[w]; b = fmaxf(b, smx[w]); }
        atomicAdd(sumSq, a);
        atomicMax(tmaxBits, __float_as_uint(b));    // valid: values >= 0
    }
}

// --------------------- main fused WMMA GEMM kernel -------------------------
// Block: 256 threads = 8 waves. Block tile: 128 rows (m) x 32 cols (k).
// Wave w: rows mBase..mBase+15, cols kBase..kBase+31 (two 16x16 WMMA tiles
// sharing one A fragment per K-step -> 8 WMMAs per A build).
__global__ __launch_bounds__(256) void frog_gemm(const float* __restrict__ pred,   // [B,2N]
                                                 const float* __restrict__ sp,     // [N,N]
                                                 const float* __restrict__ Gr,
                                                 const float* __restrict__ Gi,
                                                 float* __restrict__ rowsum,       // [B,N]
                                                 float* __restrict__ colsum,       // [B,N]
                                                 float* __restrict__ dotTm,        // [B]
                                                 float* __restrict__ sumT2) {      // [B]
    __shared__ float2 ytaS[NN];    // yta[j]
    __shared__ float2 usfS[NN];    // yta[j]*shift_factor[j]
    __shared__ float  wredD[8];
    __shared__ float  wredQ[8];

    const int b      = blockIdx.z;
    const int kBase  = blockIdx.x * 32;
    const int mGroup = blockIdx.y * 128;
    const int tid    = threadIdx.x;
    const int wave   = tid >> 5;
    const int lane   = tid & 31;
    const int lane16 = lane & 15;
    const int hi     = lane >> 4;

    // Stage pulse (and pre-phased pulse) into LDS.
    const float* pb = pred + b * (2 * NN);
    for (int j = tid; j < NN; j += 256) {
        float yr = pb[j];
        float yi = pb[NN + j];
        ytaS[j] = make_float2(yr, yi);
        float ang = -2.0f * W_CENTER_F * TS_F * (float)(j - NN / 2);
        float s, c;
        sincosf(ang, &s, &c);                       // sf = c + i*s
        usfS[j] = make_float2(yr * c - yi * s, yr * s + yi * c);
    }
    __syncthreads();

    const int mBase = mGroup + wave * 16;
    const int mA    = mBase + lane16;               // A-matrix row for this lane

    v8f arr0 = {}; v8f aii0 = {}; v8f ari0 = {}; v8f air0 = {};
    v8f arr1 = {}; v8f aii1 = {}; v8f ari1 = {}; v8f air1 = {};

    for (int j0 = 0; j0 < NN; j0 += 4) {
        const int ja = j0 + 2 * hi;                 // this lane's K pair: ja, ja+1
        // A fragment: prod[mA][ja], prod[mA][ja+1] = usf[j] * yta[(j - mA + 512) mod N]
        float2 u0 = usfS[ja];
        float2 u1 = usfS[ja + 1];
        int g0 = (ja - mA + (NN / 2)) & (NN - 1);
        int g1 = (g0 + 1) & (NN - 1);
        float2 y0 = ytaS[g0];
        float2 y1 = ytaS[g1];
        v2f ar, ai;
        ar.x = u0.x * y0.x - u0.y * y0.y;  ai.x = u0.x * y0.y + u0.y * y0.x;
        ar.y = u1.x * y1.x - u1.y * y1.y;  ai.y = u1.x * y1.y + u1.y * y1.x;
        // B fragments for the two k-tiles: G[ja..ja+1][kBase+lane16 (+16)]
        int gidx = ja * NN + kBase + lane16;
        v2f br0, bi0, br1, bi1;
        br0.x = Gr[gidx];            bi0.x = Gi[gidx];
        br0.y = Gr[gidx + NN];       bi0.y = Gi[gidx + NN];
        br1.x = Gr[gidx + 16];       bi1.x = Gi[gidx + 16];
        br1.y = Gr[gidx + NN + 16];  bi1.y = Gi[gidx + NN + 16];
        // Complex accumulate: out_r = rr - ii ; out_i = ri + ir  (per tile)
        arr0 = __builtin_amdgcn_wmma_f32_16x16x4_f32(false, ar, false, br0, (short)0, arr0, false, false);
        aii0 = __builtin_amdgcn_wmma_f32_16x16x4_f32(false, ai, false, bi0, (short)0, aii0, false, false);
        ari0 = __builtin_amdgcn_wmma_f32_16x16x4_f32(false, ar, false, bi0, (short)0, ari0, false, false);
        air0 = __builtin_amdgcn_wmma_f32_16x16x4_f32(false, ai, false, br0, (short)0, air0, false, false);
        arr1 = __builtin_amdgcn_wmma_f32_16x16x4_f32(false, ar, false, br1, (short)0, arr1, false, false);
        aii1 = __builtin_amdgcn_wmma_f32_16x16x4_f32(false, ai, false, bi1, (short)0, aii1, false, false);
        ari1 = __builtin_amdgcn_wmma_f32_16x16x4_f32(false, ar, false, bi1, (short)0, ari1, false, false);
        air1 = __builtin_amdgcn_wmma_f32_16x16x4_f32(false, ai, false, br1, (short)0, air1, false, false);
    }

    // Magnitude + fused reductions.
    // C layout: VGPR v -> (m = mBase+v+8*hi, k = kBase+lane16 [+16 for tile1]).
    float t0[8], t1[8];
    float colacc0 = 0.f, colacc1 = 0.f, dotacc = 0.f, sqacc = 0.f;
#pragma unroll
    for (int v = 0; v < 8; ++v) {
        int m = mBase + v + 8 * hi;
        const float* spRow = sp + m * NN + kBase + lane16;
        float or0 = arr0[v] - aii0[v];
        float oi0 = ari0[v] + air0[v];
        float tv0 = sqrtf(or0 * or0 + oi0 * oi0);
        t0[v] = tv0;
        colacc0 += tv0;
        dotacc  += tv0 * spRow[0];
        sqacc   += tv0 * tv0;
        float or1 = arr1[v] - aii1[v];
        float oi1 = ari1[v] + air1[v];
        float tv1 = sqrtf(or1 * or1 + oi1 * oi1);
        t1[v] = tv1;
        colacc1 += tv1;
        dotacc  += tv1 * spRow[16];
        sqacc   += tv1 * tv1;
    }
    // Row sums: butterfly over the 16 k-lanes of each half-wave (both tiles).
#pragma unroll
    for (int v = 0; v < 8; ++v) {
        float rv = t0[v] + t1[v];
        rv += __shfl_xor(rv, 1, 32);
        rv += __shfl_xor(rv, 2, 32);
        rv += __shfl_xor(rv, 4, 32);
        rv += __shfl_xor(rv, 8, 32);
        if (lane16 == 0) atomicAdd(&rowsum[b * NN + mBase + v + 8 * hi], rv);
    }
    // Column sums: combine the two half-waves (same k), one atomic per k.
    colacc0 += __shfl_xor(colacc0, 16, 32);
    colacc1 += __shfl_xor(colacc1, 16, 32);
    if (hi == 0) {
        atomicAdd(&colsum[b * NN + kBase + lane16], colacc0);
        atomicAdd(&colsum[b * NN + kBase + 16 + lane16], colacc1);
    }
    // Scalars: full wave reduce, then block reduce, one atomic each per block.
    for (int msk = 1; msk < 32; msk <<= 1) {
        dotacc += __shfl_xor(dotacc, msk, 32);
        sqacc  += __shfl_xor(sqacc,  msk, 32);
    }
    if (lane == 0) { wredD[wave] = dotacc; wredQ[wave] = sqacc; }
    __syncthreads();
    if (tid == 0) {
        float d = 0.f, q = 0.f;
        for (int w = 0; w < 8; ++w) { d += wredD[w]; q += wredQ[w]; }
        atomicAdd(&dotTm[b], d);
        atomicAdd(&sumT2[b], q);
    }
}

// ------------------------------ finalize -----------------------------------
__global__ __launch_bounds__(256) void finalize(const float* __restrict__ rowsum,
                                                const float* __restrict__ colsum,
                                                const float* __restrict__ colsumTm,
                                                const float* __restrict__ dotTm,
                                                const float* __restrict__ sumT2,
                                                const float* __restrict__ sumSq,
                                                const unsigned int* __restrict__ tmaxBits,
                                                float* __restrict__ out, int B) {
    __shared__ float redA[8];
    __shared__ float redB[8];
    int tid = threadIdx.x, lane = tid & 31, wave = tid >> 5;
    float errsum = 0.f;
    float A    = *sumSq;
    float tmax = __uint_as_float(*tmaxBits);
    float norm = (float)NN * (float)NN * tmax * tmax;
    for (int b = 0; b < B; ++b) {
        float s1 = 0.f, s2 = 0.f;
        for (int j = tid; j < NN; j += 256) {
            float rs = rowsum[b * NN + j];
            s1 += colsumTm[j] * rs;               // sum(Tmeas @ Tref)
            s2 += colsum[b * NN + j] * rs;        // sum(Tref  @ Tref)
        }
        for (int msk = 1; msk < 32; msk <<= 1) {
            s1 += __shfl_xor(s1, msk, 32);
            s2 += __shfl_xor(s2, msk, 32);
        }
        if (lane == 0) { redA[wave] = s1; redB[wave] = s2; }
        __syncthreads();
        if (tid == 0) {
            float S1 = 0.f, S2 = 0.f;
            for (int w = 0; w < 8; ++w) { S1 += redA[w]; S2 += redB[w]; }
            float mu = S1 / S2;
            float r  = A - 2.f * mu * dotTm[b] + mu * mu * sumT2[b];
            errsum += sqrtf(r / norm);
        }
        __syncthreads();
    }
    if (tid == 0) out[0] = errsum / (float)B;
}

// ------------------------------ launcher -----------------------------------
extern "C" void kernel_launch(void* const* d_in, const int* in_sizes, int n_in,
                              void* d_out, int out_size, void* d_ws, size_t ws_size,
                              hipStream_t stream) {
    const int B = in_sizes[0] / (2 * NN);           // 8
    const float* pred = (const float*)d_in[0];
    // d_in[1] (label) is unused by the returned loss.
    const float* sp   = (const float*)d_in[2];
    float* out = (float*)d_out;

    // Workspace layout (floats): Gr | Gi | rowsum | colsum | colsumTm | dotTm | sumT2 | sumSq | tmax
    float* ws       = (float*)d_ws;
    float* Gr       = ws;
    float* Gi       = Gr + (size_t)NN * NN;
    float* rowsum   = Gi + (size_t)NN * NN;
    float* colsum   = rowsum + (size_t)B * NN;
    float* colsumTm = colsum + (size_t)B * NN;
    float* dotTm    = colsumTm + NN;
    float* sumT2    = dotTm + B;
    float* sumSq    = sumT2 + B;
    unsigned int* tmaxB = (unsigned int*)(sumSq + 1);

    // 1) zero everything from rowsum through tmax (contiguous).
    int zc = 2 * B * NN + NN + 2 * B + 2;
    zero_kernel<<<(zc + 255) / 256, 256, 0, stream>>>(rowsum, zc);

    // 2) DFT matrix (fftshifts + TS folded in).
    fill_G<<<(NN * NN) / 256, 256, 0, stream>>>(Gr, Gi);

    // 3) measured-trace statistics.
    spectro_stats<<<NN / 256, 256, 0, stream>>>(sp, colsumTm, sumSq, tmaxB);

    // 4) fused complex WMMA GEMM + magnitude + reductions.
    dim3 grid(NN / 32, NN / 128, B);
    frog_gemm<<<grid, 256, 0, stream>>>(pred, sp, Gr, Gi, rowsum, colsum, dotTm, sumT2);

    // 5) scalar FROG errors -> mean.
    finalize<<<1, 256, 0, stream>>>(rowsum, colsum, colsumTm, dotTm, sumT2, sumSq, tmaxB, out, B);
}